// GraphSAGENetwork_11450382811606
// MI455X (gfx1250) — compile-verified
//
#include <hip/hip_runtime.h>
#include <hip/hip_bf16.h>

typedef __attribute__((ext_vector_type(2))) float v2f;
typedef __attribute__((ext_vector_type(8))) float v8f;

#define NODES_F 128   // feature width of every SAGE layer (F_IN == H == 128)

// ---------------------------------------------------------------------------
// Zero-fill helper (graph-capture-safe init of workspace)
// ---------------------------------------------------------------------------
__global__ void zero_f32(float* __restrict__ p, int n) {
    int i = blockIdx.x * blockDim.x + threadIdx.x;
    if (i < n) p[i] = 0.0f;
}

// ---------------------------------------------------------------------------
// Degree: deg[dst] += 1 for each edge
// ---------------------------------------------------------------------------
__global__ void compute_deg(const int* __restrict__ dst, float* __restrict__ deg, int nEdges) {
    int e = blockIdx.x * blockDim.x + threadIdx.x;
    if (e < nEdges) unsafeAtomicAdd(&deg[dst[e]], 1.0f);
}

// ---------------------------------------------------------------------------
// Edge scatter: agg[dst] += h[src]   (32 lanes per edge, float4 per lane)
// ---------------------------------------------------------------------------
__global__ void scatter_edges(const int* __restrict__ src, const int* __restrict__ dst,
                              const float* __restrict__ h, float* __restrict__ agg,
                              int nEdges) {
    long long t = (long long)blockIdx.x * blockDim.x + threadIdx.x;
    int e = (int)(t >> 5);
    int lane = (int)(t & 31);
    if (e >= nEdges) return;
    int s = src[e];
    int d = dst[e];
    const float4 v = *(const float4*)(h + (size_t)s * NODES_F + lane * 4);
    float* p = agg + (size_t)d * NODES_F + lane * 4;
    unsafeAtomicAdd(p + 0, v.x);
    unsafeAtomicAdd(p + 1, v.y);
    unsafeAtomicAdd(p + 2, v.z);
    unsafeAtomicAdd(p + 3, v.w);
}

// ---------------------------------------------------------------------------
// SAGE layer GEMM: out = (agg/deg) @ Wl + h @ Wr + b  [+ ReLU]
// One wave (32 lanes) computes one 16x16 output tile using
// V_WMMA_F32_16X16X4_F32, K = 128 in steps of 4, two WMMAs per k-step.
//
// A 16x4 f32 layout (ISA 7.12.2): lane<16 -> M=lane, VGPR0=K0,VGPR1=K1;
//                                 lane>=16 -> same M, K+2.
// B 4x16 f32 layout mirrors A: per-VGPR rows striped across lanes.
// C/D 16x16 f32: VGPR r, lanes0-15 -> M=r, lanes16-31 -> M=r+8.
// ---------------------------------------------------------------------------
__global__ __launch_bounds__(32) void sage_gemm(
    const float* __restrict__ agg, const float* __restrict__ deg,
    const float* __restrict__ h,
    const float* __restrict__ Wl, const float* __restrict__ Wr,
    const float* __restrict__ bias, float* __restrict__ out, int relu) {

    const int lane = threadIdx.x;          // 0..31
    const int m    = lane & 15;            // row within tile (A) / col (B,C)
    const int kk   = (lane >> 4) * 2;      // 0 or 2: K sub-offset for this half-wave
    const int rowA = blockIdx.x * 16 + m;
    const int colB = blockIdx.y * 16 + m;  // column owned by this lane for B and C

    const float d   = deg[rowA];
    const float inv = 1.0f / fmaxf(d, 1.0f);

    const float* aggRow = agg + (size_t)rowA * NODES_F + kk;
    const float* hRow   = h   + (size_t)rowA * NODES_F + kk;

    v8f acc = {};
    #pragma unroll
    for (int k0 = 0; k0 < NODES_F; k0 += 4) {
        // mean-aggregated neighbor features  x  Wl
        v2f a1;
        a1.x = aggRow[k0]     * inv;
        a1.y = aggRow[k0 + 1] * inv;
        v2f b1;
        b1.x = Wl[(size_t)(k0 + kk)     * NODES_F + colB];
        b1.y = Wl[(size_t)(k0 + kk + 1) * NODES_F + colB];
        acc = __builtin_amdgcn_wmma_f32_16x16x4_f32(
            false, a1, false, b1, (short)0, acc, false, false);

        // self features  x  Wr
        v2f a2;
        a2.x = hRow[k0];
        a2.y = hRow[k0 + 1];
        v2f b2;
        b2.x = Wr[(size_t)(k0 + kk)     * NODES_F + colB];
        b2.y = Wr[(size_t)(k0 + kk + 1) * NODES_F + colB];
        acc = __builtin_amdgcn_wmma_f32_16x16x4_f32(
            false, a2, false, b2, (short)0, acc, false, false);
    }

    const float bv = bias[colB];
    const int rowC = blockIdx.x * 16 + (lane >> 4) * 8;
    #pragma unroll
    for (int r = 0; r < 8; ++r) {
        float v = acc[r] + bv;
        if (relu) v = fmaxf(v, 0.0f);
        out[(size_t)(rowC + r) * NODES_F + colB] = v;
    }
}

// ---------------------------------------------------------------------------
// Global mean pool: pooled[batch[n]] += h[n]; gcount[batch[n]] += 1
// ---------------------------------------------------------------------------
__global__ void pool_nodes(const float* __restrict__ h, const int* __restrict__ batch,
                           float* __restrict__ pooled, float* __restrict__ gcount,
                           int nNodes) {
    long long t = (long long)blockIdx.x * blockDim.x + threadIdx.x;
    int node = (int)(t >> 5);
    int lane = (int)(t & 31);
    if (node >= nNodes) return;
    int g = batch[node];
    const float4 v = *(const float4*)(h + (size_t)node * NODES_F + lane * 4);
    float* p = pooled + (size_t)g * NODES_F + lane * 4;
    unsafeAtomicAdd(p + 0, v.x);
    unsafeAtomicAdd(p + 1, v.y);
    unsafeAtomicAdd(p + 2, v.z);
    unsafeAtomicAdd(p + 3, v.w);
    if (lane == 0) unsafeAtomicAdd(&gcount[g], 1.0f);
}

// ---------------------------------------------------------------------------
// MLP head + softmax: one thread per graph (64 graphs)
// z = relu(g @ W1 + b1) @ W2 + b2 ; softmax over 3 classes
// ---------------------------------------------------------------------------
__global__ __launch_bounds__(64) void head_kernel(
    const float* __restrict__ pooled, const float* __restrict__ gcount,
    const float* __restrict__ W1, const float* __restrict__ b1,
    const float* __restrict__ W2, const float* __restrict__ b2,
    float* __restrict__ out) {
    int g = threadIdx.x;           // 0..63
    if (g >= 64) return;
    const float inv = 1.0f / fmaxf(gcount[g], 1.0f);
    const float* gp = pooled + (size_t)g * NODES_F;

    float z0 = b2[0], z1 = b2[1], z2 = b2[2];
    for (int j = 0; j < 64; ++j) {
        float hsum = b1[j];
        for (int k = 0; k < NODES_F; ++k)
            hsum += gp[k] * inv * W1[(size_t)k * 64 + j];
        hsum = fmaxf(hsum, 0.0f);
        z0 += hsum * W2[(size_t)j * 3 + 0];
        z1 += hsum * W2[(size_t)j * 3 + 1];
        z2 += hsum * W2[(size_t)j * 3 + 2];
    }
    float mx = fmaxf(z0, fmaxf(z1, z2));
    float e0 = __expf(z0 - mx);
    float e1 = __expf(z1 - mx);
    float e2 = __expf(z2 - mx);
    float s = e0 + e1 + e2;
    out[g * 3 + 0] = e0 / s;
    out[g * 3 + 1] = e1 / s;
    out[g * 3 + 2] = e2 / s;
}

// ---------------------------------------------------------------------------
// Host-side launcher
// ---------------------------------------------------------------------------
extern "C" void kernel_launch(void* const* d_in, const int* in_sizes, int n_in,
                              void* d_out, int out_size, void* d_ws, size_t ws_size,
                              hipStream_t stream) {
    const float* x     = (const float*)d_in[0];
    const int*   ei    = (const int*)d_in[1];
    const int*   batch = (const int*)d_in[2];
    const float* Wl[3] = {(const float*)d_in[3], (const float*)d_in[6], (const float*)d_in[9]};
    const float* bl[3] = {(const float*)d_in[4], (const float*)d_in[7], (const float*)d_in[10]};
    const float* Wr[3] = {(const float*)d_in[5], (const float*)d_in[8], (const float*)d_in[11]};
    const float* W1 = (const float*)d_in[12];
    const float* b1 = (const float*)d_in[13];
    const float* W2 = (const float*)d_in[14];
    const float* b2 = (const float*)d_in[15];

    const int nNodes = in_sizes[0] / NODES_F;   // 50000
    const int nEdges = in_sizes[1] / 2;         // 800000
    const int* src = ei;
    const int* dst = ei + nEdges;

    // Workspace carve-up
    char* w = (char*)d_ws;
    float* agg    = (float*)w; w += (size_t)nNodes * NODES_F * sizeof(float);
    float* deg    = (float*)w; w += (size_t)nNodes * sizeof(float);
    float* hA     = (float*)w; w += (size_t)nNodes * NODES_F * sizeof(float);
    float* hB     = (float*)w; w += (size_t)nNodes * NODES_F * sizeof(float);
    float* pooled = (float*)w; w += (size_t)64 * NODES_F * sizeof(float);
    float* gcount = (float*)w; w += 64 * sizeof(float);

    const int nFeat = nNodes * NODES_F;

    // Degree (constant across layers, recomputed every call for determinism)
    zero_f32<<<(nNodes + 255) / 256, 256, 0, stream>>>(deg, nNodes);
    compute_deg<<<(nEdges + 255) / 256, 256, 0, stream>>>(dst, deg, nEdges);

    const float* hin = x;
    float* houts[3] = {hA, hB, hA};
    const long long scatterThreads = (long long)nEdges * 32;
    for (int i = 0; i < 3; ++i) {
        zero_f32<<<(nFeat + 255) / 256, 256, 0, stream>>>(agg, nFeat);
        scatter_edges<<<(unsigned)((scatterThreads + 255) / 256), 256, 0, stream>>>(
            src, dst, hin, agg, nEdges);
        dim3 grid(nNodes / 16, NODES_F / 16);   // 3125 x 8 waves
        sage_gemm<<<grid, 32, 0, stream>>>(agg, deg, hin, Wl[i], Wr[i], bl[i],
                                           houts[i], (i < 2) ? 1 : 0);
        hin = houts[i];
    }

    // pooled and gcount are contiguous -> one zero pass
    zero_f32<<<(64 * NODES_F + 64 + 255) / 256, 256, 0, stream>>>(pooled, 64 * NODES_F + 64);
    const long long poolThreads = (long long)nNodes * 32;
    pool_nodes<<<(unsigned)((poolThreads + 255) / 256), 256, 0, stream>>>(
        hin, batch, pooled, gcount, nNodes);

    head_kernel<<<1, 64, 0, stream>>>(pooled, gcount, W1, b1, W2, b2, (float*)d_out);
}